// cal_Loss_39221641347692
// MI455X (gfx1250) — compile-verified
//
#include <hip/hip_runtime.h>

// ---------------------------------------------------------------------------
// Reference:  theta = F @ G.T  (N=8192, K=16, fp32)
//   out = sum(softplus(theta*0.5)) - sum(theta where labels overlap)
//         + GAMMA * sum((B-F)^2 + (B-G)^2)
//         + ETA   * (sum(rowsum(F)^2) + sum(rowsum(G)^2))
// Strategy:
//   * V_WMMA_F32_16X16X4_F32 x4 per 16x16 tile (full-precision fp32 matmul)
//   * labels packed to 96-bit masks; "shared class" = (maskA & maskB) != 0
//   * deterministic reduction: per-block partials in d_ws + final tree sum
// ---------------------------------------------------------------------------

#define GAMMA_C 0.5f
#define ETA_C   0.5f
#define NROW    8192
#define KBITS   16
#define NLAB    81
#define NTILE1D 512                    // 8192 / 16
#define WAVES_PER_BLOCK 8
#define MAIN_BLOCKS ((NTILE1D * NTILE1D) / WAVES_PER_BLOCK)   // 32768

typedef float v2f __attribute__((ext_vector_type(2)));
typedef float v8f __attribute__((ext_vector_type(8)));

// ---- kernel 1: pack 81 x int32 {0,1} labels into 96-bit masks ---------------
__global__ void pack_labels_kernel(const int* __restrict__ img,
                                   const int* __restrict__ tex,
                                   uint4* __restrict__ imgM,
                                   uint4* __restrict__ texM) {
    int r = blockIdx.x * blockDim.x + threadIdx.x;
    if (r >= NROW) return;
    unsigned wi0 = 0, wi1 = 0, wi2 = 0, wt0 = 0, wt1 = 0, wt2 = 0;
    const int* ip = img + (long)r * NLAB;
    const int* tp = tex + (long)r * NLAB;
    for (int j = 0; j < NLAB; ++j) {
        unsigned bit = 1u << (j & 31);
        if (ip[j] != 0) { if (j < 32) wi0 |= bit; else if (j < 64) wi1 |= bit; else wi2 |= bit; }
        if (tp[j] != 0) { if (j < 32) wt0 |= bit; else if (j < 64) wt1 |= bit; else wt2 |= bit; }
    }
    imgM[r] = make_uint4(wi0, wi1, wi2, 0u);
    texM[r] = make_uint4(wt0, wt1, wt2, 0u);
}

// ---- kernel 2: one wave = one 16x16 tile of theta, fp32 WMMA ----------------
__global__ void __launch_bounds__(256)
theta_tile_kernel(const float* __restrict__ F,
                  const float* __restrict__ G,
                  const uint4* __restrict__ imgM,
                  const uint4* __restrict__ texM,
                  float* __restrict__ tilePart) {
    const int lane = threadIdx.x & 31;
    const int wave = threadIdx.x >> 5;
    const unsigned tile = blockIdx.x * WAVES_PER_BLOCK + wave;
    const int it = tile >> 9;          // / NTILE1D
    const int tt = tile & (NTILE1D - 1);
    const int i0 = it << 4;
    const int t0 = tt << 4;
    const int half = lane >> 4;        // 0: K even pair, 1: K odd pair (per ISA layout)
    const int l15  = lane & 15;

    const float* Frow = F + (long)(i0 + l15) * KBITS;   // A: rows of F
    const float* Grow = G + (long)(t0 + l15) * KBITS;   // B: rows of G (= cols of G^T)

    // A 16x4 fp32 layout: lanes 0-15 -> {K=0,K=1}, lanes 16-31 -> {K=2,K=3}
    // B  4x16 fp32 layout mirrors it with lane = N column.
    v8f c = {};
    #pragma unroll
    for (int kk = 0; kk < 4; ++kk) {
        const int kb = kk * 4 + half * 2;
        v2f a; a.x = Frow[kb]; a.y = Frow[kb + 1];
        v2f b; b.x = Grow[kb]; b.y = Grow[kb + 1];
        c = __builtin_amdgcn_wmma_f32_16x16x4_f32(
                /*neg_a=*/false, a, /*neg_b=*/false, b,
                /*c_mod=*/(short)0, c, /*reuse_a=*/false, /*reuse_b=*/false);
    }

    // C/D layout: lane = N (mod 16); VGPR r holds M = r + 8*(lane>=16)
    const uint4 tm = texM[t0 + l15];
    float part = 0.f;
    #pragma unroll
    for (int r = 0; r < 8; ++r) {
        const int row = i0 + r + half * 8;
        const uint4 im = imgM[row];
        const bool sim = (((im.x & tm.x) | (im.y & tm.y) | (im.z & tm.z)) != 0u);
        const float th = c[r];
        const float x  = 0.5f * th;
        // numerically stable softplus(x) = max(x,0) + log(1 + exp(-|x|))
        const float sp = fmaxf(x, 0.f) + __logf(1.f + __expf(-fabsf(x)));
        part += sp - (sim ? th : 0.f);
    }

    // butterfly reduce across the wave (wave32) -> identical value in all lanes
    #pragma unroll
    for (int off = 16; off > 0; off >>= 1)
        part += __shfl_xor(part, off, 32);

    __shared__ float wsum[WAVES_PER_BLOCK];
    if (lane == 0) wsum[wave] = part;
    __syncthreads();
    if (threadIdx.x == 0) {
        float s = 0.f;
        #pragma unroll
        for (int w = 0; w < WAVES_PER_BLOCK; ++w) s += wsum[w];   // fixed order
        tilePart[blockIdx.x] = s;
    }
}

// ---- kernel 3: quantization + balance terms (one thread per row) ------------
__global__ void qb_kernel(const float* __restrict__ F,
                          const float* __restrict__ G,
                          const float* __restrict__ B,
                          float* __restrict__ qbPart) {
    const int r = blockIdx.x * blockDim.x + threadIdx.x;
    float q = 0.f, sf = 0.f, sg = 0.f;
    const float* fr = F + (long)r * KBITS;
    const float* gr = G + (long)r * KBITS;
    const float* br = B + (long)r * KBITS;
    #pragma unroll
    for (int k = 0; k < KBITS; ++k) {
        const float f = fr[k], g = gr[k], b = br[k];
        const float df = b - f, dg = b - g;
        q += df * df + dg * dg;
        sf += f;
        sg += g;
    }
    float v = GAMMA_C * q + ETA_C * (sf * sf + sg * sg);

    __shared__ float s[256];
    s[threadIdx.x] = v;
    __syncthreads();
    for (int st = 128; st > 0; st >>= 1) {     // fixed-order tree
        if (threadIdx.x < st) s[threadIdx.x] += s[threadIdx.x + st];
        __syncthreads();
    }
    if (threadIdx.x == 0) qbPart[blockIdx.x] = s[0];
}

// ---- kernel 4: deterministic final reduction --------------------------------
__global__ void final_reduce_kernel(const float* __restrict__ tilePart, int nTile,
                                    const float* __restrict__ qbPart, int nQb,
                                    float* __restrict__ out) {
    __shared__ float s[256];
    float v = 0.f;
    for (int i = threadIdx.x; i < nTile; i += 256) v += tilePart[i];  // fixed order
    s[threadIdx.x] = v;
    __syncthreads();
    for (int st = 128; st > 0; st >>= 1) {
        if (threadIdx.x < st) s[threadIdx.x] += s[threadIdx.x + st];
        __syncthreads();
    }
    if (threadIdx.x == 0) {
        float total = s[0];
        for (int i = 0; i < nQb; ++i) total += qbPart[i];             // fixed order
        out[0] = total;
    }
}

// ---------------------------------------------------------------------------
extern "C" void kernel_launch(void* const* d_in, const int* in_sizes, int n_in,
                              void* d_out, int out_size, void* d_ws, size_t ws_size,
                              hipStream_t stream) {
    const float* F = (const float*)d_in[0];
    const float* G = (const float*)d_in[1];
    const float* B = (const float*)d_in[2];
    const int* img_label = (const int*)d_in[3];
    const int* tex_label = (const int*)d_in[4];
    float* out = (float*)d_out;

    // workspace layout (bytes):
    //   [0,             131072)  imgMask : 8192 x uint4
    //   [131072,        262144)  texMask : 8192 x uint4
    //   [262144,        393216)  tilePart: 32768 x float
    //   [393216,        393344)  qbPart  : 32 x float
    char* ws = (char*)d_ws;
    uint4* imgM     = (uint4*)(ws);
    uint4* texM     = (uint4*)(ws + 131072);
    float* tilePart = (float*)(ws + 262144);
    float* qbPart   = (float*)(ws + 393216);

    pack_labels_kernel<<<NROW / 256, 256, 0, stream>>>(img_label, tex_label, imgM, texM);

    theta_tile_kernel<<<MAIN_BLOCKS, 256, 0, stream>>>(F, G, imgM, texM, tilePart);

    qb_kernel<<<NROW / 256, 256, 0, stream>>>(F, G, B, qbPart);

    final_reduce_kernel<<<1, 256, 0, stream>>>(tilePart, MAIN_BLOCKS,
                                               qbPart, NROW / 256, out);
}